// GIN_37890201485516
// MI455X (gfx1250) — compile-verified
//
#include <hip/hip_runtime.h>

#define N_NODES 100000
#define N_EDGES 1600000
#define NFEAT 64
#define NHID 64
#define NCLASS 16

typedef float v2f __attribute__((ext_vector_type(2)));
typedef float v8f __attribute__((ext_vector_type(8)));

// ---------------------------------------------------------------------------
// Kernel 1: ws = (1 + eps) * x        (pure f32 stream, float4 vectorized)
// ---------------------------------------------------------------------------
__global__ __launch_bounds__(256) void gin_init(const float* __restrict__ x,
                                                const float* __restrict__ eps,
                                                float* __restrict__ agg) {
    int i = blockIdx.x * blockDim.x + threadIdx.x;   // float4 index
    const float s = 1.0f + eps[0];
    const float4* x4 = (const float4*)x;
    float4* a4 = (float4*)agg;
    if (i < (N_NODES * NFEAT) / 4) {
        float4 v = x4[i];
        v.x *= s; v.y *= s; v.z *= s; v.w *= s;
        a4[i] = v;
    }
}

// ---------------------------------------------------------------------------
// Kernel 2: scatter-add  agg[dst] += x[src]  over 1.6M edges.
// 16 threads per edge, float4 gather (row-coalesced) + 4 global_atomic_add_f32
// (no-return -> STOREcnt path). x (25.6MB) and agg (25.6MB) both stay resident
// in the 192MB L2, so the 102M atomics never touch HBM.
// ---------------------------------------------------------------------------
__global__ __launch_bounds__(256) void gin_scatter(const float* __restrict__ x,
                                                   const long long* __restrict__ ei,
                                                   float* __restrict__ agg) {
    long long gid = (long long)blockIdx.x * blockDim.x + threadIdx.x;
    int e = (int)(gid >> 4);
    int c = ((int)gid & 15) << 2;          // feature offset (0..60 step 4)
    if (e >= N_EDGES) return;
    int src = (int)ei[e];
    int dst = (int)ei[(long long)N_EDGES + e];
    float4 v = *(const float4*)(x + (size_t)src * NFEAT + c);
    float* d = agg + (size_t)dst * NFEAT + c;
    atomicAdd(d + 0, v.x);
    atomicAdd(d + 1, v.y);
    atomicAdd(d + 2, v.z);
    atomicAdd(d + 3, v.w);
}

// ---------------------------------------------------------------------------
// Kernel 3: fused MLP  out = relu(h@W1+b1)@W2+b2  via V_WMMA_F32_16X16X4_F32.
// 256 threads = 8 waves; each wave owns a 16-row tile (128 rows / block).
//
// Weights live in LDS pre-swizzled into WMMA B-fragment order so each lane's
// fragment is one contiguous ds_load_b64:
//   B frag for step kt, half p (koff=2p), col n  =  { W[(4kt+2p)*N+n], W[(4kt+2p+1)*N+n] }
//   stored at  wf[((kt*2+p)*N + n)*2 + {0,1}]
// A-fragment layout (32-bit A 16x4): lanes 0-15 -> K={0,1}, lanes 16-31 ->
// K={2,3}, M = lane&15.  C/D layout: VGPR r, lanes 0-15 -> M=r, 16-31 -> M=r+8.
// ---------------------------------------------------------------------------
__global__ __launch_bounds__(256) void gin_mlp(const float* __restrict__ agg,
                                               const float* __restrict__ W1,
                                               const float* __restrict__ b1,
                                               const float* __restrict__ W2,
                                               const float* __restrict__ b2,
                                               float* __restrict__ out) {
    __shared__ __align__(16) float w1f[64 * 64];   // fragment-ordered W1 (16KB)
    __shared__ __align__(16) float w2f[64 * 16];   // fragment-ordered W2 (4KB)
    __shared__ float b1s[64];
    __shared__ float b2s[16];
    __shared__ __align__(16) float hst[8][16 * 66]; // per-wave h1 staging

    const int tid = threadIdx.x;

    // Cooperative LDS fill, swizzling into fragment order.
    for (int l = tid; l < 64 * 64; l += 256) {
        int j  = l & 1;
        int n  = (l >> 1) & 63;
        int p  = (l >> 7) & 1;
        int kt = l >> 8;
        w1f[l] = W1[(kt * 4 + p * 2 + j) * NHID + n];
    }
    for (int l = tid; l < 64 * 16; l += 256) {
        int j  = l & 1;
        int n  = (l >> 1) & 15;
        int p  = (l >> 5) & 1;
        int kt = l >> 6;
        w2f[l] = W2[(kt * 4 + p * 2 + j) * NCLASS + n];
    }
    if (tid < 64) b1s[tid] = b1[tid];
    if (tid < 16) b2s[tid] = b2[tid];
    __syncthreads();

    const int wave = tid >> 5;
    const int lane = tid & 31;
    const int half = lane >> 4;           // 0: K pair {0,1}, 1: K pair {2,3}
    const int l16  = lane & 15;
    const int koff = half * 2;

    const int rowBase = blockIdx.x * 128 + wave * 16;

    // ---- Load A fragments for this wave's 16x64 tile of h (b64 loads) ----
    v2f afrag[16];
    const int arow = rowBase + l16;
    if (arow < N_NODES) {
        const float* ap = agg + (size_t)arow * NFEAT + koff;
        #pragma unroll
        for (int kt = 0; kt < 16; ++kt)
            afrag[kt] = *(const v2f*)(ap + kt * 4);
    } else {
        #pragma unroll
        for (int kt = 0; kt < 16; ++kt) { afrag[kt].x = 0.f; afrag[kt].y = 0.f; }
    }

    // ---- Layer 1: h1 = relu(h @ W1 + b1), 4 N-tiles x 16 K-steps ----
    #pragma unroll
    for (int nt = 0; nt < 4; ++nt) {
        v8f acc = {0.f, 0.f, 0.f, 0.f, 0.f, 0.f, 0.f, 0.f};
        #pragma unroll
        for (int kt = 0; kt < 16; ++kt) {
            v2f bfrag = *(const v2f*)&w1f[((kt * 2 + half) * 64 + nt * 16 + l16) * 2];
            acc = __builtin_amdgcn_wmma_f32_16x16x4_f32(
                false, afrag[kt], false, bfrag, (short)0, acc, false, false);
        }
        const float bb = b1s[nt * 16 + l16];
        #pragma unroll
        for (int r = 0; r < 8; ++r) {
            float v = acc[r] + bb;
            v = v > 0.f ? v : 0.f;
            hst[wave][(r + half * 8) * 66 + nt * 16 + l16] = v;   // row M, col N
        }
    }
    // Staging is wave-private: compiler-inserted s_wait_dscnt suffices,
    // no workgroup barrier needed.

    // ---- Layer 2: out = h1 @ W2 + b2, 1 N-tile x 16 K-steps ----
    v8f acc2 = {0.f, 0.f, 0.f, 0.f, 0.f, 0.f, 0.f, 0.f};
    #pragma unroll
    for (int kt = 0; kt < 16; ++kt) {
        v2f a2 = *(const v2f*)&hst[wave][l16 * 66 + kt * 4 + koff];
        v2f bf = *(const v2f*)&w2f[((kt * 2 + half) * 16 + l16) * 2];
        acc2 = __builtin_amdgcn_wmma_f32_16x16x4_f32(
            false, a2, false, bf, (short)0, acc2, false, false);
    }
    const float bb2 = b2s[l16];
    #pragma unroll
    for (int r = 0; r < 8; ++r) {
        const int row = rowBase + r + half * 8;
        if (row < N_NODES)
            out[(size_t)row * NCLASS + l16] = acc2[r] + bb2;
    }
}

// ---------------------------------------------------------------------------
extern "C" void kernel_launch(void* const* d_in, const int* in_sizes, int n_in,
                              void* d_out, int out_size, void* d_ws, size_t ws_size,
                              hipStream_t stream) {
    const float*     x   = (const float*)d_in[0];
    const long long* ei  = (const long long*)d_in[1];   // int64 edge_index [2, E]
    const float*     eps = (const float*)d_in[2];
    const float*     W1  = (const float*)d_in[3];
    const float*     b1  = (const float*)d_in[4];
    const float*     W2  = (const float*)d_in[5];
    const float*     b2  = (const float*)d_in[6];
    float* out = (float*)d_out;
    float* agg = (float*)d_ws;   // 100000*64*4 = 25.6 MB scratch accumulator

    // 1) agg = (1+eps) * x
    gin_init<<<(N_NODES * NFEAT / 4 + 255) / 256, 256, 0, stream>>>(x, eps, agg);
    // 2) agg[dst] += x[src]  (L2-resident atomics)
    gin_scatter<<<(N_EDGES * 16) / 256, 256, 0, stream>>>(x, ei, agg);
    // 3) out = relu(agg@W1+b1)@W2+b2  via f32 WMMA
    gin_mlp<<<(N_NODES + 127) / 128, 256, 0, stream>>>(agg, W1, b1, W2, b2, out);
}